// Net_81801947120042
// MI455X (gfx1250) — compile-verified
//
#include <hip/hip_runtime.h>
#include <stdint.h>

typedef __attribute__((ext_vector_type(2))) float v2f;
typedef __attribute__((ext_vector_type(8))) float v8f;

#define FOUT 64

// ---------------- degree / normalization ----------------

__global__ void deg_init_kernel(float* __restrict__ dinv, int N) {
    int i = blockIdx.x * blockDim.x + threadIdx.x;
    if (i < N) dinv[i] = 1.0f;  // self-loop contributes 1
}

__global__ void deg_count_kernel(const int* __restrict__ dst, float* __restrict__ dinv, int E) {
    int e = blockIdx.x * blockDim.x + threadIdx.x;
    if (e < E) atomicAdd(&dinv[dst[e]], 1.0f);
}

__global__ void deg_rsqrt_kernel(float* __restrict__ dinv, int N) {
    int i = blockIdx.x * blockDim.x + threadIdx.x;
    if (i < N) dinv[i] = rsqrtf(dinv[i]);  // deg >= 1 always
}

// ---------------- WMMA GEMM: Out[M,64] = A[M,K] @ W[K,64] ----------------
// Block = 256 threads = 8 waves; each wave owns a 16-row x 64-col output strip.
// V_WMMA_F32_16X16X4_F32 (full fp32 precision).
//
// W is staged global->LDS with GLOBAL_LOAD_ASYNC_TO_LDS_B128 (ASYNCcnt),
// then repacked LDS->LDS into B-fragment pair layout:
//   sPair[kp*PROW + n] = { W[2kp][n], W[2kp+1][n] }   (one ds_load_b64 per frag)
// Row stride PROW=80 pairs (640B) => consecutive kp rows are offset by 32 banks,
// so lanes 0-15 (row kp) and 16-31 (row kp+1) use disjoint bank halves.
//
// A-frag (32-bit A 16x4): lanes 0-15 rows M, VGPR pair = K=klo..klo+1, klo=2*(lane>=16).
// C/D: VGPR j holds (M=j, lanes 0-15) / (M=j+8, lanes 16-31), N = lane&15.

#define PROW 80  // pairs per kp-row in LDS (64 used + 16 pad)

template <int K>
__global__ __launch_bounds__(256) void gemm_wmma_kernel(
    const float* __restrict__ A, const float* __restrict__ W,
    float* __restrict__ Out, int M) {
    __shared__ float sStage[K * FOUT];            // verbatim copy of W
    __shared__ v2f   sPair[(K / 2) * PROW];       // pair-interleaved fragments

    // ---- async-stage W into LDS (16B per lane per issue) ----
    {
        unsigned lbase = (unsigned)(uintptr_t)&sStage[0];  // LDS byte offset
        for (int i = threadIdx.x; i < (K * FOUT) / 4; i += 256) {
            unsigned laddr = lbase + i * 16u;
            uint64_t gaddr = (uint64_t)(const void*)(W + i * 4);
            asm volatile("global_load_async_to_lds_b128 %0, %1, off"
                         :: "v"(laddr), "v"(gaddr) : "memory");
        }
        asm volatile("s_wait_asynccnt 0x0" ::: "memory");
    }
    __syncthreads();

    // ---- repack to pair layout ----
    for (int p = threadIdx.x; p < (K / 2) * FOUT; p += 256) {
        int kp = p >> 6;
        int n  = p & 63;
        v2f pr = { sStage[(2 * kp) * FOUT + n], sStage[(2 * kp + 1) * FOUT + n] };
        sPair[kp * PROW + n] = pr;
    }
    __syncthreads();

    const int wave = threadIdx.x >> 5;
    const int lane = threadIdx.x & 31;
    const int half = lane >> 4;   // which K-pair of the 4-wide step this lane holds
    const int lc   = lane & 15;

    const int m0   = blockIdx.x * 128 + wave * 16;
    int mrow  = m0 + lc;
    int mload = mrow < M ? mrow : (M - 1);  // clamp: keep EXEC all-1 for WMMA
    const float* __restrict__ Arow = A + (size_t)mload * K + half * 2;
    __builtin_prefetch(Arow, 0, 0);  // global_prefetch_b8: pull this row strip

    v8f acc0 = {0.f,0.f,0.f,0.f,0.f,0.f,0.f,0.f};
    v8f acc1 = acc0, acc2 = acc0, acc3 = acc0;

    for (int k0 = 0; k0 < K; k0 += 4) {
        v2f a = *(const v2f*)(Arow + k0);              // 8B-aligned (k0+klo even)
        const v2f* __restrict__ prow = sPair + (size_t)((k0 >> 1) + half) * PROW + lc;
        v2f b0 = prow[0];
        v2f b1 = prow[16];
        v2f b2 = prow[32];
        v2f b3 = prow[48];
        acc0 = __builtin_amdgcn_wmma_f32_16x16x4_f32(false, a, false, b0, (short)0, acc0, false, false);
        acc1 = __builtin_amdgcn_wmma_f32_16x16x4_f32(false, a, false, b1, (short)0, acc1, false, false);
        acc2 = __builtin_amdgcn_wmma_f32_16x16x4_f32(false, a, false, b2, (short)0, acc2, false, false);
        acc3 = __builtin_amdgcn_wmma_f32_16x16x4_f32(false, a, false, b3, (short)0, acc3, false, false);
    }

    const int rbase = m0 + half * 8;
    #pragma unroll
    for (int j = 0; j < 8; ++j) {
        int r = rbase + j;
        if (r < M) {
            float* __restrict__ orow = Out + (size_t)r * FOUT + lc;
            orow[0]  = acc0[j];
            orow[16] = acc1[j];
            orow[32] = acc2[j];
            orow[48] = acc3[j];
        }
    }
}

// ---------------- edge aggregation (both branches fused) ----------------
// 16 threads per edge; each thread handles 4 contiguous features.

__global__ __launch_bounds__(256) void aggregate_kernel(
    const int* __restrict__ src, const int* __restrict__ dst,
    const float* __restrict__ dinv,
    const float* __restrict__ hA, const float* __restrict__ hB,
    float* __restrict__ aggA, float* __restrict__ aggB, int E) {
    long long idx = (long long)blockIdx.x * 256 + threadIdx.x;
    int e = (int)(idx >> 4);
    if (e >= E) return;
    int t = ((int)idx & 15) * 4;

    int s = src[e], d = dst[e];
    float norm = dinv[s] * dinv[d];

    const float4 va = *(const float4*)(hA + (size_t)s * FOUT + t);
    float* __restrict__ pa = aggA + (size_t)d * FOUT + t;
    atomicAdd(pa + 0, norm * va.x);
    atomicAdd(pa + 1, norm * va.y);
    atomicAdd(pa + 2, norm * va.z);
    atomicAdd(pa + 3, norm * va.w);

    const float4 vb = *(const float4*)(hB + (size_t)s * FOUT + t);
    float* __restrict__ pb = aggB + (size_t)d * FOUT + t;
    atomicAdd(pb + 0, norm * vb.x);
    atomicAdd(pb + 1, norm * vb.y);
    atomicAdd(pb + 2, norm * vb.z);
    atomicAdd(pb + 3, norm * vb.w);
}

// ---------------- fused self-loop + bias (+ ReLU) epilogues ----------------

__global__ __launch_bounds__(256) void relu_bias_kernel(
    float* __restrict__ aggA, const float* __restrict__ hxA, const float* __restrict__ bA,
    float* __restrict__ aggB, const float* __restrict__ hxB, const float* __restrict__ bB,
    const float* __restrict__ dinv, int N) {
    long long idx = (long long)blockIdx.x * 256 + threadIdx.x;
    if (idx >= (long long)N * FOUT) return;
    int i = (int)(idx >> 6);
    int f = (int)idx & 63;
    float w = dinv[i]; w *= w;  // self-loop norm = dinv^2
    float va = aggA[idx] + w * hxA[idx] + bA[f];
    aggA[idx] = va > 0.f ? va : 0.f;
    float vb = aggB[idx] + w * hxB[idx] + bB[f];
    aggB[idx] = vb > 0.f ? vb : 0.f;
}

__global__ __launch_bounds__(256) void finalize_kernel(
    const float* __restrict__ aggA, const float* __restrict__ hxA, const float* __restrict__ bA,
    const float* __restrict__ aggB, const float* __restrict__ hxB, const float* __restrict__ bB,
    const float* __restrict__ dinv, float* __restrict__ out, int N) {
    long long idx = (long long)blockIdx.x * 256 + threadIdx.x;
    if (idx >= (long long)N * FOUT) return;
    int i = (int)(idx >> 6);
    int f = (int)idx & 63;
    float w = dinv[i]; w *= w;
    out[idx] = aggA[idx] + w * hxA[idx] + bA[f];                       // branch "in"
    out[(size_t)N * FOUT + idx] = aggB[idx] + w * hxB[idx] + bB[f];    // branch "out"
}

// ---------------- host orchestration ----------------

extern "C" void kernel_launch(void* const* d_in, const int* in_sizes, int n_in,
                              void* d_out, int out_size, void* d_ws, size_t ws_size,
                              hipStream_t stream) {
    const float* x       = (const float*)d_in[0];
    const int*   eidx    = (const int*)d_in[1];
    const float* in_W1   = (const float*)d_in[2];
    const float* in_b1   = (const float*)d_in[3];
    const float* in_W2   = (const float*)d_in[4];
    const float* in_b2   = (const float*)d_in[5];
    const float* out_W1  = (const float*)d_in[6];
    const float* out_b1  = (const float*)d_in[7];
    const float* out_W2  = (const float*)d_in[8];
    const float* out_b2  = (const float*)d_in[9];

    const int N = in_sizes[0] / 128;   // 100000
    const int E = in_sizes[1] / 2;     // 1600000
    const int* src = eidx;
    const int* dst = eidx + E;

    float* ws   = (float*)d_ws;
    float* dinv = ws;                       // N
    float* hxA  = dinv + N;                 // N*64
    float* hxB  = hxA + (size_t)N * FOUT;   // N*64
    float* aggA = hxB + (size_t)N * FOUT;   // N*64
    float* aggB = aggA + (size_t)N * FOUT;  // N*64
    const size_t featBytes = (size_t)N * FOUT * sizeof(float);

    const int TB = 256;
    const int gN   = (N + TB - 1) / TB;
    const int gE   = (E + TB - 1) / TB;
    const int gNF  = (int)(((long long)N * FOUT + TB - 1) / TB);
    const int gE16 = (int)((16LL * E + TB - 1) / TB);
    const int gGemm = (N + 127) / 128;

    // normalization coefficients
    deg_init_kernel<<<gN, TB, 0, stream>>>(dinv, N);
    deg_count_kernel<<<gE, TB, 0, stream>>>(dst, dinv, E);
    deg_rsqrt_kernel<<<gN, TB, 0, stream>>>(dinv, N);

    // ---- layer 1: hx = x @ W1 (both branches), aggregate, relu+bias ----
    gemm_wmma_kernel<128><<<gGemm, TB, 0, stream>>>(x, in_W1, hxA, N);
    gemm_wmma_kernel<128><<<gGemm, TB, 0, stream>>>(x, out_W1, hxB, N);
    hipMemsetAsync(aggA, 0, featBytes, stream);
    hipMemsetAsync(aggB, 0, featBytes, stream);
    aggregate_kernel<<<gE16, TB, 0, stream>>>(src, dst, dinv, hxA, hxB, aggA, aggB, E);
    relu_bias_kernel<<<gNF, TB, 0, stream>>>(aggA, hxA, in_b1, aggB, hxB, out_b1, dinv, N);
    // aggA/aggB now hold h1 for each branch

    // ---- layer 2: hx2 = h1 @ W2, aggregate, bias -> d_out ----
    gemm_wmma_kernel<64><<<gGemm, TB, 0, stream>>>(aggA, in_W2, hxA, N);
    gemm_wmma_kernel<64><<<gGemm, TB, 0, stream>>>(aggB, out_W2, hxB, N);
    hipMemsetAsync(aggA, 0, featBytes, stream);
    hipMemsetAsync(aggB, 0, featBytes, stream);
    aggregate_kernel<<<gE16, TB, 0, stream>>>(src, dst, dinv, hxA, hxB, aggA, aggB, E);
    finalize_kernel<<<gNF, TB, 0, stream>>>(aggA, hxA, in_b2, aggB, hxB, out_b2,
                                            dinv, (float*)d_out, N);
}